// DecoderRNN_27960237097135
// MI455X (gfx1250) — compile-verified
//
#include <hip/hip_runtime.h>
#include <math.h>

// Problem constants (match reference)
static constexpr int B = 64, R = 49, F = 512, E = 512, H = 512, V = 10000, T = 32;

typedef float v2f __attribute__((ext_vector_type(2)));
typedef float v8f __attribute__((ext_vector_type(8)));

// Up to 3 K-segments accumulated into one C:  C = sum_s A_s * W_s^T (+bias)
struct GemmSeg  { const float* A; const float* W; int lda, ldw, K; };
struct GemmSegs { GemmSeg s[3]; };

// ---------------------------------------------------------------------------
// fp32 WMMA GEMM (V_WMMA_F32_16X16X4_F32), one wave -> 16x64 strip of C.
// Ping-pong register double buffering over K-chunks of 8 (loop step 16):
// each half issues the other buffer's loads before consuming its own, so
// loads overlap WMMA issue and no rotation copies are needed.
// EXEC stays all-ones everywhere (wave-uniform exits, clamped N-edge
// addressing, store-side masking only).
// Fragment layouts per CDNA5 ISA:
//   A 16x4 f32: lane L (r=L&15, hi=L>>4) holds A[r, k+2hi .. k+2hi+1]
//   B 4x16 f32: lane L holds W[col, k+2hi .. k+2hi+1]   (NT: both K-contig)
//   C 16x16 f32: vgpr v, lane L -> C[row0 + v + 8hi, col0 + (L&15)]
// ---------------------------------------------------------------------------
__global__ __launch_bounds__(128) void gemm_f32_wmma(
    GemmSegs segs, int nseg,
    float* __restrict__ C, int ldc, int M, int N,
    const float* __restrict__ bias0, const float* __restrict__ bias1)
{
    const int wave  = threadIdx.x >> 5;
    const int lane  = threadIdx.x & 31;
    const int gwave = blockIdx.x * 4 + wave;

    const int mtiles  = M >> 4;            // M is a multiple of 16 here
    const int ngroups = (N + 63) >> 6;     // groups of 4 16-wide tiles
    const int mt = gwave % mtiles;
    const int ng = gwave / mtiles;
    if (ng >= ngroups) return;             // wave-uniform

    const int row0 = mt << 4;
    const int col0 = ng << 6;
    const int r    = lane & 15;
    const int hi   = lane >> 4;

    v8f acc[4] = {v8f{}, v8f{}, v8f{}, v8f{}};

#pragma unroll 1
    for (int s = 0; s < nseg; ++s) {
        const GemmSeg sg = segs.s[s];
        const float* arow = sg.A + (size_t)(row0 + r) * sg.lda + 2 * hi;
        const float* wrow[4];
#pragma unroll
        for (int j = 0; j < 4; ++j) {
            int cidx = col0 + j * 16 + r;
            if (cidx >= N) cidx = N - 1;   // clamp: keep EXEC all-ones
            wrow[j] = sg.W + (size_t)cidx * sg.ldw + 2 * hi;
        }
        const int K = sg.K;                // multiple of 16

        auto load_frag = [&](int k, v2f& a0, v2f& a1, v2f (&b0)[4], v2f (&b1)[4]) {
            a0 = *(const v2f*)(arow + k);
            a1 = *(const v2f*)(arow + k + 4);
#pragma unroll
            for (int j = 0; j < 4; ++j) {
                b0[j] = *(const v2f*)(wrow[j] + k);
                b1[j] = *(const v2f*)(wrow[j] + k + 4);
            }
        };
        auto do_wmma = [&](v2f a0, v2f a1, v2f (&b0)[4], v2f (&b1)[4]) {
#pragma unroll
            for (int j = 0; j < 4; ++j)
                acc[j] = __builtin_amdgcn_wmma_f32_16x16x4_f32(
                    false, a0, false, b0[j], (short)0, acc[j], false, false);
#pragma unroll
            for (int j = 0; j < 4; ++j)
                acc[j] = __builtin_amdgcn_wmma_f32_16x16x4_f32(
                    false, a1, false, b1[j], (short)0, acc[j], false, false);
        };

        // ping-pong buffers
        v2f a0A, a1A, b0A[4], b1A[4];
        v2f a0B, a1B, b0B[4], b1B[4];
        load_frag(0, a0A, a1A, b0A, b1A);

#pragma unroll 1
        for (int k = 0; k < K; k += 16) {
            // prefetch chunk k+8 into B while consuming A
            load_frag(k + 8, a0B, a1B, b0B, b1B);
            do_wmma(a0A, a1A, b0A, b1A);
            // prefetch chunk k+16 into A while consuming B
            int kn = k + 16;
            if (kn > K - 8) kn = K - 8;    // clamped in-bounds redundant reload
            load_frag(kn, a0A, a1A, b0A, b1A);
            do_wmma(a0B, a1B, b0B, b1B);
        }
    }

#pragma unroll
    for (int j = 0; j < 4; ++j) {
        const int cidx = col0 + j * 16 + r;
        if (cidx < N) {
            float bias = 0.f;
            if (bias0) bias += bias0[cidx];
            if (bias1) bias += bias1[cidx];
#pragma unroll
            for (int v = 0; v < 8; ++v) {
                const int    rr  = row0 + v + 8 * hi;
                const size_t idx = (size_t)rr * ldc + cidx;
                C[idx] = acc[j][v] + bias;
            }
        }
    }
}

static inline void launch_gemm1(const float* A, int lda, const float* W, int ldw,
                                float* C, int ldc, int M, int N, int K,
                                const float* bias0, const float* bias1,
                                hipStream_t stream)
{
    GemmSegs segs{};
    segs.s[0] = {A, W, lda, ldw, K};
    const int waves  = (M >> 4) * ((N + 63) >> 6);
    const int blocks = (waves + 3) >> 2;
    gemm_f32_wmma<<<blocks, 128, 0, stream>>>(segs, 1, C, ldc, M, N, bias0, bias1);
}

// ---------------------------------------------------------------------------
// avg[b,f] = mean over R regions of features[b,r,f]
// ---------------------------------------------------------------------------
__global__ void avg_pool_kernel(const float* __restrict__ feats,
                                float* __restrict__ avg)
{
    int i = blockIdx.x * blockDim.x + threadIdx.x;
    if (i >= B * F) return;
    const int b = i >> 9, f = i & 511;
    const float* p = feats + (size_t)b * R * F + f;
    float s = 0.f;
#pragma unroll 7
    for (int rr = 0; rr < R; ++rr) s += p[rr * F];
    avg[i] = s * (1.0f / (float)R);
}

// ---------------------------------------------------------------------------
// Fused additive attention, one block (8 wave32s) per batch element:
//   e[r]   = tanh(enc_att[b,r,:] + dec_att[b,:]) . v_w + v_b
//   alpha  = softmax(e) ; ctx[f] = sum_r alpha[r] * features[b,r,f]
// Regions striped across the 8 waves; intra-wave shuffle reduction.
// ---------------------------------------------------------------------------
__global__ __launch_bounds__(256) void attention_kernel(
    const float* __restrict__ enc_att, const float* __restrict__ dec_att,
    const float* __restrict__ v_w, const float* __restrict__ v_b,
    const float* __restrict__ feats, float* __restrict__ context)
{
    __shared__ float dec_s[H];
    __shared__ float alpha_s[64];

    const int b    = blockIdx.x;
    const int tid  = threadIdx.x;
    const int wave = tid >> 5;
    const int lane = tid & 31;

    for (int h = tid; h < H; h += 256) dec_s[h] = dec_att[b * H + h];
    __syncthreads();

    const float* enc_b = enc_att + (size_t)b * R * H;
    // each wave handles regions wave, wave+8, ... ; lanes stripe over H
    for (int rgn = wave; rgn < R; rgn += 8) {
        float p = 0.f;
        const float* e_r = enc_b + rgn * H;
#pragma unroll 4
        for (int h = lane; h < H; h += 32)
            p += tanhf(e_r[h] + dec_s[h]) * v_w[h];
#pragma unroll
        for (int off = 16; off > 0; off >>= 1)
            p += __shfl_down(p, off, 32);
        if (lane == 0) alpha_s[rgn] = p + v_b[0];
    }
    __syncthreads();

    if (tid == 0) {
        float mx = alpha_s[0];
        for (int rgn = 1; rgn < R; ++rgn) mx = fmaxf(mx, alpha_s[rgn]);
        float sum = 0.f;
        for (int rgn = 0; rgn < R; ++rgn) {
            float ex = __expf(alpha_s[rgn] - mx);
            alpha_s[rgn] = ex;
            sum += ex;
        }
        const float inv = 1.0f / sum;
        for (int rgn = 0; rgn < R; ++rgn) alpha_s[rgn] *= inv;
    }
    __syncthreads();

    const float* fb = feats + (size_t)b * R * F;
    for (int f = tid; f < F; f += 256) {
        float s = 0.f;
#pragma unroll 7
        for (int rgn = 0; rgn < R; ++rgn) s += alpha_s[rgn] * fb[rgn * F + f];
        context[b * F + f] = s;
    }
}

// ---------------------------------------------------------------------------
// LSTM pointwise: gates[b, 0:H]=i, [H:2H]=f, [2H:3H]=g, [3H:4H]=o
// ---------------------------------------------------------------------------
__device__ __forceinline__ float sigmoidf_(float x) {
    return 1.0f / (1.0f + __expf(-x));
}

__global__ void lstm_pointwise_kernel(const float* __restrict__ gates,
                                      float* __restrict__ h,
                                      float* __restrict__ c)
{
    int i = blockIdx.x * blockDim.x + threadIdx.x;
    if (i >= B * H) return;
    const int b = i >> 9, j = i & 511;
    const float* g = gates + (size_t)b * 4 * H;
    const float ig = sigmoidf_(g[j]);
    const float fg = sigmoidf_(g[H + j]);
    const float gg = tanhf(g[2 * H + j]);
    const float og = sigmoidf_(g[3 * H + j]);
    const float cn = fg * c[i] + ig * gg;
    c[i] = cn;
    h[i] = og * tanhf(cn);
}

// ---------------------------------------------------------------------------
// Host launcher
// ---------------------------------------------------------------------------
extern "C" void kernel_launch(void* const* d_in, const int* in_sizes, int n_in,
                              void* d_out, int out_size, void* d_ws, size_t ws_size,
                              hipStream_t stream)
{
    (void)in_sizes; (void)n_in; (void)out_size; (void)ws_size;
    const float* features = (const float*)d_in[0];
    const float* captions = (const float*)d_in[1];
    const float* W_enc    = (const float*)d_in[2];
    const float* b_enc    = (const float*)d_in[3];
    const float* W_dec    = (const float*)d_in[4];
    const float* b_dec    = (const float*)d_in[5];
    const float* v_w      = (const float*)d_in[6];
    const float* v_b      = (const float*)d_in[7];
    const float* W_ih     = (const float*)d_in[8];
    const float* W_hh     = (const float*)d_in[9];
    const float* b_ih     = (const float*)d_in[10];
    const float* b_hh     = (const float*)d_in[11];
    const float* W_init_h = (const float*)d_in[12];
    const float* b_init_h = (const float*)d_in[13];
    const float* W_init_c = (const float*)d_in[14];
    const float* b_init_c = (const float*)d_in[15];
    const float* W_out    = (const float*)d_in[16];
    const float* b_out    = (const float*)d_in[17];
    float* out = (float*)d_out;

    // Workspace layout (fp32)
    float* ws      = (float*)d_ws;
    float* enc_att = ws;                               // B*R*H
    float* avg     = enc_att + (size_t)B * R * H;      // B*F
    float* h       = avg + (size_t)B * F;              // B*H
    float* c       = h + (size_t)B * H;                // B*H
    float* dec_att = c + (size_t)B * H;                // B*H
    float* context = dec_att + (size_t)B * H;          // B*F
    float* gates   = context + (size_t)B * F;          // B*4H

    // ---- hoisted, h-independent work ----
    avg_pool_kernel<<<(B * F + 255) / 256, 256, 0, stream>>>(features, avg);
    launch_gemm1(avg, F, W_init_h, F, h, H, B, H, F, b_init_h, nullptr, stream);
    launch_gemm1(avg, F, W_init_c, F, c, H, B, H, F, b_init_c, nullptr, stream);
    // enc_att[b,r,:] = features[b,r,:] @ W_enc^T + b_enc  -> one (B*R)xHxF GEMM
    launch_gemm1(features, F, W_enc, F, enc_att, H, B * R, H, F, b_enc, nullptr, stream);

    // ---- sequential decode: 31 steps, 5 launches per step ----
    for (int t = 0; t < T - 1; ++t) {
        // dec_att = h @ W_dec^T + b_dec
        launch_gemm1(h, H, W_dec, H, dec_att, H, B, H, H, b_dec, nullptr, stream);
        // attention -> context
        attention_kernel<<<B, 256, 0, stream>>>(enc_att, dec_att, v_w, v_b,
                                                features, context);
        // gates = [embed|context] @ W_ih^T + h @ W_hh^T + (b_ih + b_hh)
        // three K-segments fused into ONE kernel (register accumulation)
        {
            GemmSegs segs{};
            segs.s[0] = {captions + (size_t)t * E, W_ih,     T * E, E + F, E};
            segs.s[1] = {context,                  W_ih + E, F,     E + F, F};
            segs.s[2] = {h,                        W_hh,     H,     H,     H};
            const int waves  = (B >> 4) * ((4 * H + 63) >> 6);
            const int blocks = (waves + 3) >> 2;
            gemm_f32_wmma<<<blocks, 128, 0, stream>>>(segs, 3, gates, 4 * H,
                                                      B, 4 * H, b_ih, b_hh);
        }
        // LSTM cell update (in place on h, c)
        lstm_pointwise_kernel<<<(B * H + 255) / 256, 256, 0, stream>>>(gates, h, c);
        // logits = h @ W_out^T + b_out, written directly into out[:, t, :]
        launch_gemm1(h, H, W_out, H, out + (size_t)t * V, (T - 1) * V,
                     B, V, H, b_out, nullptr, stream);
    }
}